// YOLOLoss_81750407512127
// MI455X (gfx1250) — compile-verified
//
#include <hip/hip_runtime.h>
#include <cstdint>
#include <cstddef>

typedef float vfloat4 __attribute__((ext_vector_type(4)));

namespace {
constexpr int   kNM            = 3;
constexpr int   kAttrs         = 85;          // 5 + 80 classes
constexpr int   kH             = 76;
constexpr int   kW             = 76;
constexpr int   kHW            = kH * kW;     // 5776
constexpr int   kBS            = 16;
constexpr int   kPlanes        = kBS * kNM;   // 48
constexpr int   kCells         = kPlanes * kHW;      // 277248
constexpr int   kGrpPerPlane   = kHW / 4;            // 1444
constexpr int   kBlock         = 192;                // 6 waves of 32
constexpr int   kCellsPerBlock = kBlock * 4;         // 768
constexpr int   kBlocks        = kCells / kCellsPerBlock; // 361 exact
constexpr float kValConf       = 0.1f;
constexpr float kInvW          = 1.0f / (float)kW;
constexpr float kInvH          = 1.0f / (float)kH;
} // namespace

__device__ __forceinline__ float sigmoid_f(float x) {
  return 1.0f / (1.0f + __expf(-x));
}

__global__ __launch_bounds__(kBlock) void yolo_head_decode(
    const float* __restrict__ x,        // [48, 85, 5776]
    const float* __restrict__ anchors,  // [3, 2] scaled
    float* __restrict__ out)            // pred [kCells*7] ++ valid [kCells]
{
  // Staging buffer: 768 cells * 7 floats = 5376 floats = 21504 bytes.
  __shared__ float lds_pred[kCellsPerBlock * 7];

  const int t  = threadIdx.x;
  const int g  = blockIdx.x * kBlock + t;      // 4-cell group index, no tail
  const int p  = g / kGrpPerPlane;             // plane = b*3 + m
  const int s4 = g - p * kGrpPerPlane;
  const int s0 = s4 * 4;                       // cell offset in plane, %4==0
  const int h  = s0 / kW;
  const int w0 = s0 - h * kW;                  // 76%4==0 -> w0..w0+3 in-row
  const int m  = p - (p / kNM) * kNM;

  const float aW = anchors[2 * m + 0];
  const float aH = anchors[2 * m + 1];

  const float* plane = x + (size_t)p * (size_t)(kAttrs * kHW) + s0;

  // --- streaming NT loads: one aligned b128 per attribute plane per lane ---
  vfloat4 tx = __builtin_nontemporal_load((const vfloat4*)(plane + 0 * kHW));
  vfloat4 ty = __builtin_nontemporal_load((const vfloat4*)(plane + 1 * kHW));
  vfloat4 tw = __builtin_nontemporal_load((const vfloat4*)(plane + 2 * kHW));
  vfloat4 th = __builtin_nontemporal_load((const vfloat4*)(plane + 3 * kHW));
  vfloat4 tc = __builtin_nontemporal_load((const vfloat4*)(plane + 4 * kHW));

  vfloat4 best = __builtin_nontemporal_load((const vfloat4*)(plane + 5 * kHW));
  vfloat4 bidx = {0.0f, 0.0f, 0.0f, 0.0f};

#pragma unroll 4
  for (int a = 6; a < kAttrs; ++a) {
    vfloat4 v = __builtin_nontemporal_load((const vfloat4*)(plane + a * kHW));
    const float fa = (float)(a - 5);
#pragma unroll
    for (int j = 0; j < 4; ++j) {
      if (v[j] > best[j]) { best[j] = v[j]; bidx[j] = fa; }  // first-max kept
    }
  }

  vfloat4 validv;
#pragma unroll
  for (int j = 0; j < 4; ++j) {
    const float cx   = (sigmoid_f(tx[j]) + (float)(w0 + j)) * kInvW;
    const float cy   = (sigmoid_f(ty[j]) + (float)h) * kInvH;
    const float bw   = __expf(tw[j]) * aW;
    const float bh   = __expf(th[j]) * aH;
    const float conf = sigmoid_f(tc[j]);
    const float cs   = sigmoid_f(best[j]);
    const float vf   = (conf > kValConf) ? 1.0f : 0.0f;

    float* dst = &lds_pred[(t * 4 + j) * 7];   // stride-7 dwords: bank-clean
    dst[0] = (cx - 0.5f * bw) * vf;
    dst[1] = (cy - 0.5f * bh) * vf;
    dst[2] = (cx + 0.5f * bw) * vf;
    dst[3] = (cy + 0.5f * bh) * vf;
    dst[4] = conf * vf;
    dst[5] = cs * vf;
    dst[6] = bidx[j] * vf;
    validv[j] = vf;
  }

  // valid mask: already coalesced -> direct NT b128 store
  __builtin_nontemporal_store(
      validv, (vfloat4*)(out + (size_t)kCells * 7 + (size_t)g * 4));

  __syncthreads();

  // --- drain pred LDS -> HBM via CDNA5 async LDS-to-memory DMA path ---
  // 5376 floats = 1344 b128 chunks = 7 per thread; block's pred region is a
  // single contiguous span, so consecutive lanes store consecutive 16B.
  const uint32_t lds_base =
      (uint32_t)(uintptr_t)&lds_pred[0];       // low 32 bits == LDS byte addr
  const uint64_t gbase =
      (uint64_t)(uintptr_t)out +
      (uint64_t)blockIdx.x * (uint64_t)(kCellsPerBlock * 7 * sizeof(float));

#pragma unroll
  for (int i = 0; i < 7; ++i) {
    const uint32_t off  = (uint32_t)((t + i * kBlock) * 16);
    const uint32_t lpos = lds_base + off;
    // GVS form: mem = SADDR + VADDR[31:0]; vdata = LDS byte address.
    asm volatile("global_store_async_from_lds_b128 %0, %1, %2"
                 :
                 : "v"(off), "v"(lpos), "s"(gbase)
                 : "memory");
  }
  asm volatile("s_wait_asynccnt 0" ::: "memory");
}

extern "C" void kernel_launch(void* const* d_in, const int* in_sizes, int n_in,
                              void* d_out, int out_size, void* d_ws,
                              size_t ws_size, hipStream_t stream) {
  (void)in_sizes; (void)n_in; (void)d_ws; (void)ws_size; (void)out_size;
  const float* x       = (const float*)d_in[0];
  const float* anchors = (const float*)d_in[1];
  float*       out     = (float*)d_out;
  yolo_head_decode<<<dim3(kBlocks), dim3(kBlock), 0, stream>>>(x, anchors, out);
}